// RelGCNCov_17575006175421
// MI455X (gfx1250) — compile-verified
//
#include <hip/hip_runtime.h>
#include <hip/hip_bf16.h>
#include <math.h>

typedef __attribute__((ext_vector_type(2))) float v2f;
typedef __attribute__((ext_vector_type(8))) float v8f;

#define D 200
#define DPAD 208   // 13*16, N padded for WMMA tiling

__device__ __forceinline__ void atom_add_f32(float* p, float v) {
    __hip_atomic_fetch_add(p, v, __ATOMIC_RELAXED, __HIP_MEMORY_SCOPE_AGENT);
}

// ---------------------------------------------------------------- zero stats
__global__ void k_zero(float* __restrict__ p, int n) {
    int t = blockIdx.x * blockDim.x + threadIdx.x;
    if (t < n) p[t] = 0.f;
}

// --------------------------------------------- WpPk = (1/3) * R @ loop_w
// R[j,k] = loop_rel[(j+k)%200].  Stored K-pair interleaved:
//   WpPk[((j>>1)*DPAD + n)*2 + (j&1)] = Wp[j][n]   (cols >= 200 zero)
// so a B fragment {row k+2hi, row k+1+2hi} at col n is ONE b64 load.
__global__ void k_wp(const float* __restrict__ loop_rel,
                     const float* __restrict__ loop_w,
                     float* __restrict__ WpPk) {
    int t = blockIdx.x * blockDim.x + threadIdx.x;
    if (t >= D * DPAD) return;
    int j = t / DPAD, n = t % DPAD;
    float s = 0.f;
    if (n < D) {
        for (int k = 0; k < D; ++k) {
            int jk = j + k; if (jk >= D) jk -= D;
            s = fmaf(loop_rel[jk], loop_w[k * D + n], s);
        }
        s *= (1.f / 3.f);
    }
    WpPk[(size_t)((j >> 1) * DPAD + n) * 2 + (j & 1)] = s;
}

// --------------- Min/Mout = softmax(rel_repr @ {in_w,out_w}, dim=1) / 3
// and out2 = rel_repr @ w_rel (second reference output, no softmax)
__global__ void k_msgrel(const float* __restrict__ rel_repr,
                         const float* __restrict__ in_w,
                         const float* __restrict__ out_w,
                         const float* __restrict__ w_rel,
                         float* __restrict__ Min, float* __restrict__ Mout,
                         float* __restrict__ out2, int R2) {
    int b = blockIdx.x / R2;       // 0: in_w->Min, 1: out_w->Mout, 2: w_rel->out2
    int r = blockIdx.x % R2;
    int t = threadIdx.x;           // 256 threads, first 200 produce a column
    const float* W = (b == 0) ? in_w : (b == 1) ? out_w : w_rel;
    float val = 0.f;
    if (t < D) {
        const float* hrow = rel_repr + (size_t)r * D;
        for (int k = 0; k < D; ++k) val = fmaf(hrow[k], W[k * D + t], val);
    }
    if (b == 2) { if (t < D) out2[(size_t)r * D + t] = val; return; }

    __shared__ float red[256];
    red[t] = (t < D) ? val : -3.4e38f;
    __syncthreads();
    for (int s = 128; s > 0; s >>= 1) {
        if (t < s) red[t] = fmaxf(red[t], red[t + s]);
        __syncthreads();
    }
    float mx = red[0];
    __syncthreads();
    float ex = (t < D) ? expf(val - mx) : 0.f;
    red[t] = ex;
    __syncthreads();
    for (int s = 128; s > 0; s >>= 1) {
        if (t < s) red[t] += red[t + s];
        __syncthreads();
    }
    float sc = (1.f / 3.f) / red[0];
    if (t < D) ((b == 0) ? Min : Mout)[(size_t)r * D + t] = ex * sc;
}

// -------------------------- out = x @ Wp + bias  via V_WMMA_F32_16X16X4_F32
// blockDim = 64 (2 waves); each wave computes one 16-row x 208-col strip.
__global__ void k_loop_gemm(const float* __restrict__ x,
                            const float* __restrict__ WpPk,  // K-pair packed
                            const float* __restrict__ bias,
                            float* __restrict__ out, int V) {
    const int wave = threadIdx.x >> 5;
    const int lane = threadIdx.x & 31;
    const int tile = blockIdx.x * 2 + wave;
    const int m0 = tile * 16;
    if (m0 >= V) return;                 // wave-uniform: EXEC stays all-ones
    const int lr = lane & 15;            // M row (A) / N col (B,D) within tile
    const int hi = lane >> 4;            // K-pair select per ISA f32 layouts

    v8f acc[13] = {};
    const float* arow = x + (size_t)(m0 + lr) * D;
    const v2f* bbase = reinterpret_cast<const v2f*>(WpPk);

    for (int k = 0; k < D; k += 4) {
        // A 16x4 frag: lanes0-15 hold K=k,k+1; lanes16-31 hold K=k+2,k+3
        v2f a = *reinterpret_cast<const v2f*>(arow + k + 2 * hi);
        const v2f* brow = bbase + (size_t)((k >> 1) + hi) * DPAD;
        #pragma unroll
        for (int nt = 0; nt < 13; ++nt) {
            // B 4x16 frag: one b64 load = {row k+2hi, row k+1+2hi} at col n
            v2f b = brow[nt * 16 + lr];
            acc[nt] = __builtin_amdgcn_wmma_f32_16x16x4_f32(
                false, a, false, b, (short)0, acc[nt], false, false);
        }
    }
    // D layout: VGPR r holds row (m0 + 8*hi + r), col = nt*16 + lr
    #pragma unroll
    for (int nt = 0; nt < 13; ++nt) {
        const int n = nt * 16 + lr;
        if (n < D) {
            const float bv = bias[n];
            #pragma unroll
            for (int r = 0; r < 8; ++r)
                out[(size_t)(m0 + hi * 8 + r) * D + n] = acc[nt][r] + bv;
        }
    }
}

// ---------------- scatter: out[dst[e]] += edge_norm[e] * M_sel[edge_type[e]]
// one thread per (edge, 4-feature chunk): 50 chunks per edge
__global__ void k_scatter(const float* __restrict__ Min,
                          const float* __restrict__ Mout,
                          const float* __restrict__ edge_norm,
                          const int* __restrict__ edge_type,
                          const int* __restrict__ dst,
                          float* __restrict__ out, int E, int half) {
    long long tid = (long long)blockIdx.x * blockDim.x + threadIdx.x;
    long long e = tid / 50;
    int c = (int)(tid % 50);
    if (e >= E) return;
    float w = edge_norm[e];
    int r = edge_type[e];
    const float* Mrow = ((e < half) ? Min : Mout) + (size_t)r * D + c * 4;
    float* orow = out + (size_t)dst[e] * D + c * 4;
    #pragma unroll
    for (int i = 0; i < 4; ++i) atom_add_f32(orow + i, w * Mrow[i]);
}

// ------------------------------------ BN: column sums / sums of squares
__global__ void k_bnstats(const float* __restrict__ out,
                          float* __restrict__ stats, int V) {
    int n = threadIdx.x;                 // 256 threads, first 200 used
    if (n >= D) return;
    float s = 0.f, q = 0.f;
    for (int v = blockIdx.x; v < V; v += gridDim.x) {
        float val = out[(size_t)v * D + n];
        s += val; q = fmaf(val, val, q);
    }
    atom_add_f32(&stats[n], s);
    atom_add_f32(&stats[D + n], q);
}

__global__ void k_bnapply(float* __restrict__ out,
                          const float* __restrict__ stats,
                          const float* __restrict__ bnw,
                          const float* __restrict__ bnb, int V) {
    long long t = (long long)blockIdx.x * blockDim.x + threadIdx.x;
    if (t >= (long long)V * D) return;
    int n = (int)(t % D);
    float invV = 1.f / (float)V;
    float mean = stats[n] * invV;
    float var = fmaf(-mean, mean, stats[D + n] * invV);
    out[t] = (out[t] - mean) * rsqrtf(var + 1e-5f) * bnw[n] + bnb[n];
}

extern "C" void kernel_launch(void* const* d_in, const int* in_sizes, int n_in,
                              void* d_out, int out_size, void* d_ws, size_t ws_size,
                              hipStream_t stream) {
    const float* x         = (const float*)d_in[0];
    const float* rel_repr  = (const float*)d_in[1];
    const float* edge_norm = (const float*)d_in[2];
    const float* in_w      = (const float*)d_in[3];
    const float* out_w     = (const float*)d_in[4];
    const float* loop_w    = (const float*)d_in[5];
    const float* w_rel     = (const float*)d_in[6];
    const float* loop_rel  = (const float*)d_in[7];
    const float* bias      = (const float*)d_in[8];
    const float* bn_w      = (const float*)d_in[9];
    const float* bn_b      = (const float*)d_in[10];
    const int*   edge_type = (const int*)d_in[11];
    const int*   dst       = (const int*)d_in[12];
    float* out = (float*)d_out;

    const int V  = in_sizes[0] / D;      // 100000
    const int R2 = in_sizes[1] / D;      // 474
    const int E  = in_sizes[2];          // 1000000
    const int half = E / 2;

    float* ws    = (float*)d_ws;
    float* Min   = ws;                          // R2*D
    float* Mout  = Min  + (size_t)R2 * D;       // R2*D
    float* WpPk  = Mout + (size_t)R2 * D;       // D*DPAD (K-pair packed)
    float* stats = WpPk + (size_t)D * DPAD;     // 2*D
    float* out2  = out  + (size_t)V * D;        // second reference output

    k_zero<<<2, 256, 0, stream>>>(stats, 2 * D);
    k_wp<<<(D * DPAD + 255) / 256, 256, 0, stream>>>(loop_rel, loop_w, WpPk);
    k_msgrel<<<R2 * 3, 256, 0, stream>>>(rel_repr, in_w, out_w, w_rel,
                                         Min, Mout, out2, R2);

    int mtiles = (V + 15) / 16;                           // 6250
    k_loop_gemm<<<(mtiles + 1) / 2, 64, 0, stream>>>(x, WpPk, bias, out, V);

    long long sth = (long long)E * 50;
    k_scatter<<<(unsigned)((sth + 255) / 256), 256, 0, stream>>>(
        Min, Mout, edge_norm, edge_type, dst, out, E, half);

    k_bnstats<<<200, 256, 0, stream>>>(out, stats, V);

    long long tot = (long long)V * D;
    k_bnapply<<<(unsigned)((tot + 255) / 256), 256, 0, stream>>>(
        out, stats, bn_w, bn_b, V);
}